// GraphEncoder_2276332667294
// MI455X (gfx1250) — compile-verified
//
#include <hip/hip_runtime.h>
#include <hip/hip_bf16.h>
#include <math.h>

// ---------------------------------------------------------------------------
// GATv2 graph encoder for MI455X (gfx1250, wave32).
// GEMMs use V_WMMA_F32_16X16X4_F32 (fp32 WMMA). All GEMM M dims are multiples
// of 16 -> no boundary predication. Weights are pre-transposed so both A and
// B fragments are contiguous global_load_b64's.
// ---------------------------------------------------------------------------

#define N_NODES 20000
#define N_EDGES 320000
#define N_GRAPHS 64
#define HC 128      // H*C
#define HEADS 8
#define CPH 16
#define HID 64
#define EPS 1e-5f
#define SLOPE 0.2f

typedef __attribute__((ext_vector_type(2))) float v2f;
typedef __attribute__((ext_vector_type(8))) float v8f;

// --------------------------- weight transpose ------------------------------
// Wt[n*K + k] = W[k*N + n]   (tiny: <=16384 elements, L2 resident)
__global__ void transpose_w(const float* __restrict__ W, float* __restrict__ Wt,
                            int K, int N)
{
    int t = blockIdx.x * blockDim.x + threadIdx.x;
    if (t >= K * N) return;
    int k = t / N, n = t - k * N;
    Wt[(size_t)n * K + k] = W[t];
}

// ---------------------------------------------------------------------------
// fp32 WMMA GEMM:  Y[M,N] = X[M,K] @ W[K,N] + bias (opt relu)
// Requires M % 16 == 0.  Wt is W transposed (N x K).
// One wave -> one 16x16 tile. blockDim.x = 32*(N/16), blockIdx.x = m-tile.
// A frag (16x4 f32): lane r=lane&15 holds row; half=lane>>4 selects K pair
//   a = {X[m][k+2h], X[m][k+2h+1]}          (contiguous -> b64)
// B frag: lane holds col n; b = {W[k+2h][n], W[k+2h+1][n]} = Wt[n][k+2h..+1]
// D: lane holds col n = lane&15; VGPR j holds row (8*half + j).
// ---------------------------------------------------------------------------
__global__ void gemm_wmma_f32(const float* __restrict__ X,
                              const float* __restrict__ Wt,
                              const float* __restrict__ bias,
                              float* __restrict__ Y,
                              int M, int K, int N, int relu)
{
    int wave = threadIdx.x >> 5;
    int lane = threadIdx.x & 31;
    int half = lane >> 4;
    int r    = lane & 15;
    int mt   = blockIdx.x * 16;
    int nt   = wave * 16;

    const float* xp = X  + (size_t)(mt + r) * K + 2 * half;
    const float* wp = Wt + (size_t)(nt + r) * K + 2 * half;

    float bv = bias ? bias[nt + r] : 0.0f;
    v8f acc = { bv, bv, bv, bv, bv, bv, bv, bv };

#pragma unroll 8
    for (int k = 0; k < K; k += 4) {
        v2f a = *(const v2f*)(xp + k);
        v2f b = *(const v2f*)(wp + k);
        acc = __builtin_amdgcn_wmma_f32_16x16x4_f32(
                  false, a, false, b, (short)0, acc, false, false);
    }

    float* yp = Y + (size_t)(mt + half * 8) * N + nt + r;
#pragma unroll
    for (int j = 0; j < 8; ++j) {
        float v = acc[j];
        if (relu) v = v > 0.0f ? v : 0.0f;
        yp[(size_t)j * N] = v;
    }
}

// --------------------------- self-loop mean attr ---------------------------
__global__ void loop_accum(const int* __restrict__ dst, const float* __restrict__ ea,
                           float* __restrict__ sums, float* __restrict__ cnt, int E)
{
    int e = blockIdx.x * blockDim.x + threadIdx.x;
    if (e >= E) return;
    int d = dst[e];
    atomicAdd(&sums[d], ea[e]);
    atomicAdd(&cnt[d], 1.0f);
}

__global__ void loop_fin(const float* __restrict__ sums, const float* __restrict__ cnt,
                         float* __restrict__ attr, int Nn)
{
    int i = blockIdx.x * blockDim.x + threadIdx.x;
    if (i >= Nn) return;
    float c = cnt[i];
    attr[i] = c > 0.0f ? sums[i] / c : 0.0f;
}

// --------------------------- edge attention passes -------------------------
__device__ __forceinline__ unsigned enc_f32(float f) {
    unsigned b = __float_as_uint(f);
    return b ^ (((int)b >> 31) | 0x80000000u);
}
__device__ __forceinline__ float dec_f32(unsigned k) {
    unsigned b = (k & 0x80000000u) ? (k ^ 0x80000000u) : ~k;
    return __uint_as_float(b);
}

// pass 1: s[e,h] = sum_c att[h,c]*lrelu(xl[src]+xr[dst]+attr*We);  segment max
__global__ void edge_score(const float* __restrict__ xl, const float* __restrict__ xr,
                           const int* __restrict__ src, const int* __restrict__ dst,
                           const float* __restrict__ ea, const float* __restrict__ lattr,
                           const float* __restrict__ We, const float* __restrict__ att,
                           float* __restrict__ s_buf, unsigned* __restrict__ smax,
                           int E, int Nn)
{
    int t = blockIdx.x * blockDim.x + threadIdx.x;
    if (t >= (E + Nn) * HEADS) return;
    int e = t >> 3, h = t & 7;
    int si, di; float attr;
    if (e < E) { si = src[e]; di = dst[e]; attr = ea[e]; }
    else       { si = di = e - E;          attr = lattr[si]; }

    const float4* pl = (const float4*)(xl + (size_t)si * HC + h * CPH);
    const float4* pr = (const float4*)(xr + (size_t)di * HC + h * CPH);
    const float4* pw = (const float4*)(We + h * CPH);
    const float4* pa = (const float4*)(att + h * CPH);
    float s = 0.0f;
#pragma unroll
    for (int q = 0; q < 4; ++q) {
        float4 l4 = pl[q], r4 = pr[q], w4 = pw[q], a4 = pa[q];
        float m;
        m = l4.x + r4.x + attr * w4.x; m = m > 0.0f ? m : SLOPE * m; s += m * a4.x;
        m = l4.y + r4.y + attr * w4.y; m = m > 0.0f ? m : SLOPE * m; s += m * a4.y;
        m = l4.z + r4.z + attr * w4.z; m = m > 0.0f ? m : SLOPE * m; s += m * a4.z;
        m = l4.w + r4.w + attr * w4.w; m = m > 0.0f ? m : SLOPE * m; s += m * a4.w;
    }
    s_buf[t] = s;
    atomicMax(&smax[di * HEADS + h], enc_f32(s));
}

// pass 2: a = exp(s - smax[dst]);  denom[dst] += a
__global__ void edge_exp(const float* __restrict__ s_buf, const unsigned* __restrict__ smax,
                         const int* __restrict__ dst, float* __restrict__ a_buf,
                         float* __restrict__ denom, int E, int Nn)
{
    int t = blockIdx.x * blockDim.x + threadIdx.x;
    if (t >= (E + Nn) * HEADS) return;
    int e = t >> 3, h = t & 7;
    int d = (e < E) ? dst[e] : (e - E);
    float a = __expf(s_buf[t] - dec_f32(smax[d * HEADS + h]));
    a_buf[t] = a;
    atomicAdd(&denom[d * HEADS + h], a);
}

// pass 3: out[dst,ch] += (a/denom) * xl[src,ch]
__global__ void edge_aggr(const float* __restrict__ xl, const int* __restrict__ src,
                          const int* __restrict__ dst, const float* __restrict__ a_buf,
                          const float* __restrict__ denom, float* __restrict__ gat,
                          int E, int Nn)
{
    int t = blockIdx.x * blockDim.x + threadIdx.x;
    if (t >= (E + Nn) * HC) return;
    int e = t >> 7, ch = t & 127, h = ch >> 4;
    int si, di;
    if (e < E) { si = src[e]; di = dst[e]; }
    else       { si = di = e - E; }
    float alpha = a_buf[e * HEADS + h] / denom[di * HEADS + h];
    atomicAdd(&gat[(size_t)di * HC + ch], alpha * xl[(size_t)si * HC + ch]);
}

// ------------------------------- batchnorm ---------------------------------
__global__ void bn_stats(const float* __restrict__ X, const float* __restrict__ bias,
                         float* __restrict__ sum, float* __restrict__ sumsq, int M)
{
    int ch = threadIdx.x;           // blockDim = 128
    float s = 0.0f, ss = 0.0f;
    for (int r = blockIdx.x; r < M; r += gridDim.x) {
        float v = X[(size_t)r * HC + ch] + bias[ch];
        s += v; ss += v * v;
    }
    atomicAdd(&sum[ch], s);
    atomicAdd(&sumsq[ch], ss);
}

__global__ void bn_finalize(const float* __restrict__ sum, const float* __restrict__ sumsq,
                            float* __restrict__ mu, float* __restrict__ inv, int M)
{
    int ch = threadIdx.x;           // 128 threads
    float m = sum[ch] / (float)M;
    float var = sumsq[ch] / (float)M - m * m;
    mu[ch] = m;
    inv[ch] = rsqrtf(var + EPS);
}

__global__ void bn_apply(const float* __restrict__ X, const float* __restrict__ bias,
                         const float* __restrict__ gamma, const float* __restrict__ beta,
                         const float* __restrict__ mu, const float* __restrict__ inv,
                         const float* __restrict__ res, float* __restrict__ Y, int M)
{
    int t = blockIdx.x * blockDim.x + threadIdx.x;
    if (t >= M * HC) return;
    int ch = t & 127;
    float v = X[t] + bias[ch];
    v = gamma[ch] * (v - mu[ch]) * inv[ch] + beta[ch];
    if (res) v += res[t];
    Y[t] = v > 0.0f ? v : 0.0f;
}

// ------------------------------- pooling -----------------------------------
__global__ void pool_accum(const float* __restrict__ h, const int* __restrict__ batch,
                           float* __restrict__ gsum, float* __restrict__ gcnt, int M)
{
    int t = blockIdx.x * blockDim.x + threadIdx.x;
    if (t >= M * HC) return;
    int r = t >> 7, ch = t & 127;
    int g = batch[r];
    atomicAdd(&gsum[g * HC + ch], h[t]);
    if (ch == 0) atomicAdd(&gcnt[g], 1.0f);
}

__global__ void pool_fin(const float* __restrict__ gsum, const float* __restrict__ gcnt,
                         float* __restrict__ gmean)
{
    int t = blockIdx.x * blockDim.x + threadIdx.x;
    if (t >= N_GRAPHS * HC) return;
    float c = gcnt[t >> 7];
    gmean[t] = gsum[t] / (c > 1.0f ? c : 1.0f);
}

// ---------------------------------------------------------------------------
extern "C" void kernel_launch(void* const* d_in, const int* in_sizes, int n_in,
                              void* d_out, int out_size, void* d_ws, size_t ws_size,
                              hipStream_t stream)
{
    const int Nn = N_NODES, E = N_EDGES, ET = N_EDGES + N_NODES;

    // ---- inputs (setup_inputs dict order; params = sorted pytree leaves) ---
    const float* x   = (const float*)d_in[0];
    const int*   src = (const int*)d_in[1];
    const int*   dst = src + E;
    const float* ea  = (const float*)d_in[2];
    const int*   bat = (const int*)d_in[3];
    // params: W0, Wp, b0, bp, then l0..l3 each {We,Wl,Wr,att,beta,bias,bl,br,gamma}
    const float* W0 = (const float*)d_in[4];
    const float* Wp = (const float*)d_in[5];
    const float* b0 = (const float*)d_in[6];
    const float* bp = (const float*)d_in[7];

    // ---- workspace carve-up (floats) --------------------------------------
    float* w = (float*)d_ws;
    size_t off = 0;
    auto grab = [&](size_t n) { float* p = w + off; off += (n + 63) & ~(size_t)63; return p; };
    float*    hA     = grab((size_t)Nn * HC);
    float*    hB     = grab((size_t)Nn * HC);
    float*    xl     = grab((size_t)Nn * HC);
    float*    xr     = grab((size_t)Nn * HC);
    float*    gat    = grab((size_t)Nn * HC);
    float*    s_buf  = grab((size_t)ET * HEADS);
    float*    a_buf  = grab((size_t)ET * HEADS);
    unsigned* smax   = (unsigned*)grab((size_t)Nn * HEADS);
    float*    denom  = grab((size_t)Nn * HEADS);
    float*    lsums  = grab(Nn);
    float*    lcnt   = grab(Nn);
    float*    lattr  = grab(Nn);
    float*    bn_sum = grab(128);
    float*    bn_ssq = grab(128);
    float*    bn_mu  = grab(128);
    float*    bn_inv = grab(128);
    float*    gsum   = grab(N_GRAPHS * HC);
    float*    gcnt   = grab(N_GRAPHS);
    float*    gmean  = grab(N_GRAPHS * HC);
    float*    WtA    = grab((size_t)HC * HC);   // transposed Wl / W0 / Wp
    float*    WtB    = grab((size_t)HC * HC);   // transposed Wr
    (void)ws_size; (void)n_in; (void)in_sizes; (void)out_size;

    const int T = 256;
    auto blk = [](long long n, int t) { return (unsigned)((n + t - 1) / t); };

    // ---- self-loop attrs (PyG fill_value='mean') --------------------------
    hipMemsetAsync(lsums, 0, sizeof(float) * 2 * ((Nn + 63) & ~63), stream); // lsums+lcnt
    loop_accum<<<blk(E, T), T, 0, stream>>>(dst, ea, lsums, lcnt, E);
    loop_fin<<<blk(Nn, T), T, 0, stream>>>(lsums, lcnt, lattr, Nn);

    // ---- node embedding: h0 = relu(x @ W0 + b0), K=4 (single WMMA step) ---
    transpose_w<<<blk(4 * HID, T), T, 0, stream>>>(W0, WtA, 4, HID);
    gemm_wmma_f32<<<blk(Nn, 16), 32 * (HID / 16), 0, stream>>>(x, WtA, b0, hA, Nn, 4, HID, 1);

    // ---- 4 GATv2 layers ---------------------------------------------------
    float* hin = hA; float* hout = hB; int Kin = HID;
    for (int l = 0; l < 4; ++l) {
        int base = 8 + 9 * l;
        const float* We    = (const float*)d_in[base + 0];
        const float* Wl    = (const float*)d_in[base + 1];
        const float* Wr    = (const float*)d_in[base + 2];
        const float* att   = (const float*)d_in[base + 3];
        const float* beta  = (const float*)d_in[base + 4];
        const float* bias  = (const float*)d_in[base + 5];
        const float* bl    = (const float*)d_in[base + 6];
        const float* br    = (const float*)d_in[base + 7];
        const float* gamma = (const float*)d_in[base + 8];

        hipMemsetAsync(smax,  0, sizeof(unsigned) * Nn * HEADS, stream);
        hipMemsetAsync(denom, 0, sizeof(float) * Nn * HEADS, stream);
        hipMemsetAsync(gat,   0, sizeof(float) * (size_t)Nn * HC, stream);
        hipMemsetAsync(bn_sum, 0, sizeof(float) * 128, stream);
        hipMemsetAsync(bn_ssq, 0, sizeof(float) * 128, stream);

        transpose_w<<<blk(Kin * HC, T), T, 0, stream>>>(Wl, WtA, Kin, HC);
        transpose_w<<<blk(Kin * HC, T), T, 0, stream>>>(Wr, WtB, Kin, HC);
        gemm_wmma_f32<<<blk(Nn, 16), 32 * (HC / 16), 0, stream>>>(hin, WtA, bl, xl, Nn, Kin, HC, 0);
        gemm_wmma_f32<<<blk(Nn, 16), 32 * (HC / 16), 0, stream>>>(hin, WtB, br, xr, Nn, Kin, HC, 0);

        edge_score<<<blk((long long)ET * HEADS, T), T, 0, stream>>>(
            xl, xr, src, dst, ea, lattr, We, att, s_buf, smax, E, Nn);
        edge_exp<<<blk((long long)ET * HEADS, T), T, 0, stream>>>(
            s_buf, smax, dst, a_buf, denom, E, Nn);
        edge_aggr<<<blk((long long)ET * HC, T), T, 0, stream>>>(
            xl, src, dst, a_buf, denom, gat, E, Nn);

        bn_stats<<<256, 128, 0, stream>>>(gat, bias, bn_sum, bn_ssq, Nn);
        bn_finalize<<<1, 128, 0, stream>>>(bn_sum, bn_ssq, bn_mu, bn_inv, Nn);
        bn_apply<<<blk((long long)Nn * HC, T), T, 0, stream>>>(
            gat, bias, gamma, beta, bn_mu, bn_inv, (l > 0) ? hin : nullptr, hout, Nn);

        float* tmp = hin; hin = hout; hout = tmp;
        Kin = HC;
    }

    // ---- global mean pool + final projection ------------------------------
    hipMemsetAsync(gsum, 0, sizeof(float) * N_GRAPHS * HC, stream);
    hipMemsetAsync(gcnt, 0, sizeof(float) * N_GRAPHS, stream);
    pool_accum<<<blk((long long)Nn * HC, T), T, 0, stream>>>(hin, bat, gsum, gcnt, Nn);
    pool_fin<<<blk(N_GRAPHS * HC, T), T, 0, stream>>>(gsum, gcnt, gmean);

    transpose_w<<<blk(HC * HC, T), T, 0, stream>>>(Wp, WtA, HC, HC);
    gemm_wmma_f32<<<blk(N_GRAPHS, 16), 32 * (HC / 16), 0, stream>>>(
        gmean, WtA, bp, (float*)d_out, N_GRAPHS, HC, HC, 1);
}